// QuantNorm1D_new_52424370815662
// MI455X (gfx1250) — compile-verified
//
#include <hip/hip_runtime.h>
#include <hip/hip_bf16.h>
#include <cstdint>

// Problem constants (from reference)
#define NSAMP   8192
#define NPRE    8128
#define BATCH   64
#define FEAT    512
#define BW_N    0.16493848884661177f   // 8192^(-0.2)
#define SPLIT   4                      // split-K over the sample dimension
#define SCHUNK  (NSAMP / SPLIT)        // 2048 samples per partial
#define OUT_ELEMS (BATCH * FEAT)       // 32768

// ---------------------------------------------------------------------------
// Branchless erf (Abramowitz-Stegun 7.1.26, |err| <= 1.5e-7).
// rcp + 5 FMA + exp + copysign; trans ops co-issue with FMA on CDNA5.
// ---------------------------------------------------------------------------
__device__ __forceinline__ float fast_erf(float x) {
    float ax = fabsf(x);
    float t  = __builtin_amdgcn_rcpf(fmaf(0.3275911f, ax, 1.0f));
    float p  = fmaf(1.061405429f, t, -1.453152027f);
    p = fmaf(p, t, 1.421413741f);
    p = fmaf(p, t, -0.284496736f);
    p = fmaf(p, t, 0.254829592f);
    p = p * t;
    float e = __expf(-ax * ax);             // underflows to 0 for |x|>~9.3
    float y = fmaf(-p, e, 1.0f);
    return copysignf(y, x);
}

// ---------------------------------------------------------------------------
// Kernel A: per-column stats. One block (256 thr) per column f.
//  - stages monotonic uint keys of the column (prefill ++ x) in LDS
//  - double sum/sumsq partials -> ws
//  - exact 3-level radix select (11/11/10 bits) for ranks
//    {2047, 2048, 6143, 6144}  ->  iqr[f]
// ---------------------------------------------------------------------------
__global__ __launch_bounds__(256) void col_stats_kernel(
    const float* __restrict__ x, const float* __restrict__ pre,
    double* __restrict__ psum, double* __restrict__ psq,
    float* __restrict__ iqr)
{
    __shared__ unsigned skey[NSAMP];    // 32 KB
    __shared__ unsigned hist[2048];     // 8 KB
    __shared__ unsigned part[256];
    __shared__ double   red[256];
    __shared__ unsigned selBin, selRank;

    const int tid = threadIdx.x;
    const int f   = blockIdx.x;

    double ts = 0.0, tq = 0.0;
    for (int i = tid; i < NSAMP; i += 256) {
        float v = (i < NPRE) ? pre[(size_t)i * FEAT + f]
                             : x[(size_t)(i - NPRE) * FEAT + f];
        unsigned u = __float_as_uint(v);
        skey[i] = (u & 0x80000000u) ? ~u : (u | 0x80000000u);  // order-preserving
        ts += (double)v;
        tq += (double)v * (double)v;
    }
    __syncthreads();

    // block reductions (sum, sumsq) -> ws partials
    red[tid] = ts; __syncthreads();
    for (int off = 128; off > 0; off >>= 1) {
        if (tid < off) red[tid] += red[tid + off];
        __syncthreads();
    }
    if (tid == 0) psum[f] = red[0];
    __syncthreads();
    red[tid] = tq; __syncthreads();
    for (int off = 128; off > 0; off >>= 1) {
        if (tid < off) red[tid] += red[tid + off];
        __syncthreads();
    }
    if (tid == 0) psq[f] = red[0];
    __syncthreads();

    // ---- exact radix select for the 4 needed order statistics ----
    const unsigned ranks[4] = {2047u, 2048u, 6143u, 6144u};
    float q[4];
    for (int ri = 0; ri < 4; ++ri) {
        unsigned prefix = 0u, pmask = 0u, lrank = ranks[ri];
        for (int lev = 0; lev < 3; ++lev) {
            const int      shift = (lev == 0) ? 21 : (lev == 1) ? 10 : 0;
            const unsigned nb    = (lev == 2) ? 1024u : 2048u;
            const int      bpt   = (int)(nb / 256u);          // bins per thread

            for (unsigned bN = tid; bN < nb; bN += 256) hist[bN] = 0u;
            __syncthreads();
            for (int i = tid; i < NSAMP; i += 256) {
                unsigned key = skey[i];
                if ((key & pmask) == prefix)
                    atomicAdd(&hist[(key >> shift) & (nb - 1u)], 1u);
            }
            __syncthreads();

            // per-thread partial, then Hillis-Steele inclusive scan of 256
            unsigned lsum = 0u;
            for (int jb = 0; jb < bpt; ++jb) lsum += hist[tid * bpt + jb];
            part[tid] = lsum;
            __syncthreads();
            for (int off = 1; off < 256; off <<= 1) {
                unsigned v2 = (tid >= off) ? part[tid - off] : 0u;
                __syncthreads();
                part[tid] += v2;
                __syncthreads();
            }
            unsigned incl = part[tid];
            unsigned excl = incl - lsum;
            if (excl <= lrank && lrank < incl) {   // unique crossing thread
                unsigned c = excl;
                for (int jb = 0; jb < bpt; ++jb) {
                    unsigned h = hist[tid * bpt + jb];
                    if (lrank < c + h) { selBin = (unsigned)(tid * bpt + jb);
                                         selRank = lrank - c; break; }
                    c += h;
                }
            }
            __syncthreads();
            prefix |= (selBin << shift);
            pmask  |= (nb - 1u) << shift;
            lrank   = selRank;
            __syncthreads();
        }
        // prefix is the exact key of the ranks[ri]-th smallest value
        unsigned u = (prefix & 0x80000000u) ? (prefix & 0x7FFFFFFFu) : ~prefix;
        q[ri] = __uint_as_float(u);
    }

    if (tid == 0) {
        // jnp.quantile 'linear': pos 0.25*8191=2047.75, 0.75*8191=6143.25
        float q25 = q[0] + 0.75f * (q[1] - q[0]);
        float q75 = q[2] + 0.25f * (q[3] - q[2]);
        iqr[f] = q75 - q25;
    }
}

// ---------------------------------------------------------------------------
// Kernel B: scalar std over all elements + per-column bandwidth. 1 block.
// ---------------------------------------------------------------------------
__global__ __launch_bounds__(512) void bandwidth_kernel(
    const double* __restrict__ psum, const double* __restrict__ psq,
    const float* __restrict__ iqr, float* __restrict__ bw)
{
    __shared__ double rs[512];
    __shared__ double rq[512];
    __shared__ float  s_std;

    const int t = threadIdx.x;
    rs[t] = psum[t];
    rq[t] = psq[t];
    __syncthreads();
    for (int off = 256; off > 0; off >>= 1) {
        if (t < off) { rs[t] += rs[t + off]; rq[t] += rq[t + off]; }
        __syncthreads();
    }
    if (t == 0) {
        const double M = (double)NSAMP * (double)FEAT;     // 4194304
        double var = (rq[0] - rs[0] * rs[0] / M) / (M - 1.0);
        s_std = (float)sqrt(var > 0.0 ? var : 0.0);
    }
    __syncthreads();
    bw[t] = 0.9f * fminf(s_std, iqr[t] * (1.0f / 1.34f)) * BW_N;
}

// ---------------------------------------------------------------------------
// Kernel C: KDE partial sums. Grid (FEAT/4, SPLIT), 256 threads (8 wave32s).
// Block (fx, y) owns 4 columns x 64 batch rows over samples
// [y*2048, (y+1)*2048). Tile staged COLUMN-MAJOR in LDS via 4 async
// global->LDS B32 copies per thread per chunk (CDNA5 ASYNCcnt path),
// double buffered; compute reads wave-broadcast ds_load_b128.
// ---------------------------------------------------------------------------
#define CH 256   // rows per chunk

__global__ __launch_bounds__(256) void kde_partial_kernel(
    const float* __restrict__ x, const float* __restrict__ pre,
    const float* __restrict__ bw, float* __restrict__ partial)
{
    __shared__ alignas(16) float buf[2][4 * CH];   // [phase][col*CH + row], 2x4KB

    const int tid   = threadIdx.x;
    const int f0    = blockIdx.x * 4;
    const int sbase = blockIdx.y * SCHUNK;
    const int col   = tid >> 6;        // 0..3 (uniform within each wave32)
    const int b     = tid & 63;
    const int f     = f0 + col;

    const float xv = x[(size_t)b * FEAT + f];
    const float m  = (1.0f / bw[f]) * 0.70710678118654752f;  // 1/(bw*sqrt(2))
    const float a  = xv * m;

    // thread t copies row (sbase+c*CH+t), cols f0..f0+3, into col-major LDS
    auto issue = [&](int c, int phase) {
        int s = sbase + c * CH + tid;
        const float* g = (s < NPRE)
            ? (pre + (size_t)s * FEAT + f0)
            : (x + (size_t)(s - NPRE) * FEAT + f0);
        // low 32 bits of the flat pointer == wave-relative LDS byte offset
        unsigned l0 = (unsigned)(uintptr_t)(&buf[phase][tid]);
        asm volatile("global_load_async_to_lds_b32 %0, %1, off"
                     :: "v"(l0),                   "v"(g)     : "memory");
        asm volatile("global_load_async_to_lds_b32 %0, %1, off"
                     :: "v"(l0 + 1u * CH * 4u),    "v"(g + 1) : "memory");
        asm volatile("global_load_async_to_lds_b32 %0, %1, off"
                     :: "v"(l0 + 2u * CH * 4u),    "v"(g + 2) : "memory");
        asm volatile("global_load_async_to_lds_b32 %0, %1, off"
                     :: "v"(l0 + 3u * CH * 4u),    "v"(g + 3) : "memory");
    };

    issue(0, 0);

    double acc = 0.0;
    const int NCHUNK = SCHUNK / CH;    // 8
    for (int c = 0; c < NCHUNK; ++c) {
        const int phase = c & 1;
        if (c + 1 < NCHUNK) {
            issue(c + 1, phase ^ 1);
            // 4 prefetch copies stay in flight; chunk c's 4 are complete
            asm volatile("s_wait_asynccnt 4" ::: "memory");
        } else {
            asm volatile("s_wait_asynccnt 0" ::: "memory");
        }
        __syncthreads();   // all 8 waves' slices of chunk c are in LDS

        const float* lb = &buf[phase][col * CH];   // wave-uniform base
        float cs0 = 0.0f, cs1 = 0.0f, cs2 = 0.0f, cs3 = 0.0f;
        #pragma unroll 2
        for (int r = 0; r < CH; r += 8) {
            float4 dA = *(const float4*)(lb + r);       // ds_load_b128 broadcast
            float4 dB = *(const float4*)(lb + r + 4);
            cs0 += fast_erf(fmaf(-dA.x, m, a));
            cs1 += fast_erf(fmaf(-dA.y, m, a));
            cs2 += fast_erf(fmaf(-dA.z, m, a));
            cs3 += fast_erf(fmaf(-dA.w, m, a));
            cs0 += fast_erf(fmaf(-dB.x, m, a));
            cs1 += fast_erf(fmaf(-dB.y, m, a));
            cs2 += fast_erf(fmaf(-dB.z, m, a));
            cs3 += fast_erf(fmaf(-dB.w, m, a));
        }
        acc += (double)((cs0 + cs1) + (cs2 + cs3));
        __syncthreads();   // done reading buf[phase] before it is overwritten
    }

    partial[(size_t)blockIdx.y * OUT_ELEMS + (size_t)b * FEAT + f] = (float)acc;
}

// ---------------------------------------------------------------------------
// Kernel D: combine split-K partials. out = 0.5 + sum_erf/16384
// ---------------------------------------------------------------------------
__global__ __launch_bounds__(256) void kde_combine_kernel(
    const float* __restrict__ partial, float* __restrict__ out)
{
    int i = blockIdx.x * 256 + threadIdx.x;
    float s = 0.0f;
    #pragma unroll
    for (int y = 0; y < SPLIT; ++y)
        s += partial[(size_t)y * OUT_ELEMS + i];
    out[i] = 0.5f + s * (1.0f / 16384.0f);
}

// ---------------------------------------------------------------------------
extern "C" void kernel_launch(void* const* d_in, const int* in_sizes, int n_in,
                              void* d_out, int out_size, void* d_ws, size_t ws_size,
                              hipStream_t stream) {
    const float* x   = (const float*)d_in[0];   // [64, 512]
    const float* pre = (const float*)d_in[1];   // [8128, 512]
    float* out = (float*)d_out;                 // [64, 512]

    // workspace layout (~524 KB)
    double* psum    = (double*)d_ws;            // [512]
    double* psq     = psum + FEAT;              // [512]
    float*  iqr     = (float*)(psq + FEAT);     // [512]
    float*  bw      = iqr + FEAT;               // [512]
    float*  partial = bw + FEAT;                // [SPLIT * 32768]

    col_stats_kernel<<<FEAT, 256, 0, stream>>>(x, pre, psum, psq, iqr);
    bandwidth_kernel<<<1, 512, 0, stream>>>(psum, psq, iqr, bw);
    dim3 grid(FEAT / 4, SPLIT);
    kde_partial_kernel<<<grid, 256, 0, stream>>>(x, pre, bw, partial);
    kde_combine_kernel<<<OUT_ELEMS / 256, 256, 0, stream>>>(partial, out);
}